// GATLayer_22265110462672
// MI455X (gfx1250) — compile-verified
//
#include <hip/hip_runtime.h>
#include <hip/hip_bf16.h>

typedef __attribute__((ext_vector_type(2))) float v2f;
typedef __attribute__((ext_vector_type(8))) float v8f;

#define IN_DIM  128
#define OUT_DIM 64

// ---------------------------------------------------------------------------
// Kernel 0: zero the output accumulator (float4 vectorized, grid-stride)
// ---------------------------------------------------------------------------
__global__ void gat_zero_out(float4* __restrict__ out, int n4) {
    int i = blockIdx.x * blockDim.x + threadIdx.x;
    int stride = gridDim.x * blockDim.x;
    float4 z4 = make_float4(0.f, 0.f, 0.f, 0.f);
    for (; i < n4; i += stride) out[i] = z4;
}

// ---------------------------------------------------------------------------
// Kernel 1: z = h @ W_fc using V_WMMA_F32_16X16X4_F32
// One wave computes a 16-row x 64-col tile (4 accumulators of 16x16 f32).
// K-loop: 128 / 4 = 32 iterations, 4 WMMAs each.
// ---------------------------------------------------------------------------
__global__ void gat_gemm_z(const float* __restrict__ h,
                           const float* __restrict__ W,   // [128,64] row-major
                           float* __restrict__ z,
                           int nNodes) {
    const int wave = (blockIdx.x * blockDim.x + threadIdx.x) >> 5;
    const int lane = threadIdx.x & 31;
    const int row0 = wave * 16;
    if (row0 >= nNodes) return;

    const int half = lane >> 4;    // 0: lanes 0-15, 1: lanes 16-31
    const int l16  = lane & 15;

    // A-matrix addressing (16x4 f32, MxK):
    //   lanes 0-15  hold {A[M][k+0], A[M][k+1]},  M = lane
    //   lanes 16-31 hold {A[M][k+2], A[M][k+3]},  M = lane-16
    const float* hrow = h + (size_t)(row0 + l16) * IN_DIM + half * 2;

    v8f acc0 = {}, acc1 = {}, acc2 = {}, acc3 = {};

    for (int k = 0; k < IN_DIM; k += 4) {
        v2f a;
        a.x = hrow[k];
        a.y = hrow[k + 1];

        // B-matrix addressing (4x16 f32, KxN), col tiles at n0 = 0,16,32,48:
        //   lanes 0-15:  VGPR0 = B[k+0][n], VGPR1 = B[k+1][n]
        //   lanes 16-31: VGPR0 = B[k+2][n], VGPR1 = B[k+3][n]
        const float* wk = W + (size_t)(k + half * 2) * OUT_DIM + l16;
        v2f b0, b1, b2, b3;
        b0.x = wk[0];       b0.y = wk[OUT_DIM + 0];
        b1.x = wk[16];      b1.y = wk[OUT_DIM + 16];
        b2.x = wk[32];      b2.y = wk[OUT_DIM + 32];
        b3.x = wk[48];      b3.y = wk[OUT_DIM + 48];

        // 8 args: (neg_a, A, neg_b, B, c_mod, C, reuse_a, reuse_b)
        acc0 = __builtin_amdgcn_wmma_f32_16x16x4_f32(false, a, false, b0,
                                                     (short)0, acc0, false, false);
        acc1 = __builtin_amdgcn_wmma_f32_16x16x4_f32(false, a, false, b1,
                                                     (short)0, acc1, false, false);
        acc2 = __builtin_amdgcn_wmma_f32_16x16x4_f32(false, a, false, b2,
                                                     (short)0, acc2, false, false);
        acc3 = __builtin_amdgcn_wmma_f32_16x16x4_f32(false, a, false, b3,
                                                     (short)0, acc3, false, false);
    }

    // D-matrix store (16x16 f32 C/D layout):
    //   lanes 0-15:  VGPR v = D[row0 + v     ][n0 + lane]
    //   lanes 16-31: VGPR v = D[row0 + v + 8 ][n0 + lane-16]
    float* zt = z + (size_t)row0 * OUT_DIM + l16;
#pragma unroll
    for (int v = 0; v < 8; ++v) {
        const size_t r = (size_t)(v + 8 * half) * OUT_DIM;
        zt[r + 0]  = acc0[v];
        zt[r + 16] = acc1[v];
        zt[r + 32] = acc2[v];
        zt[r + 48] = acc3[v];
    }
}

// ---------------------------------------------------------------------------
// Kernel 2: per-edge attention + sigmoid-weighted scatter-add.
// One wave32 per edge; each lane owns output dims {lane, lane+32}.
// ---------------------------------------------------------------------------
__global__ void gat_edge(const float* __restrict__ z,
                         const int* __restrict__ src,
                         const int* __restrict__ dst,
                         const float* __restrict__ a_attn,  // [128]
                         float* __restrict__ out,
                         int nEdges) {
    const int edge = (blockIdx.x * blockDim.x + threadIdx.x) >> 5;
    const int lane = threadIdx.x & 31;
    if (edge >= nEdges) return;

    const int s = src[edge];
    const int d = dst[edge];
    const float* zs = z + (size_t)s * OUT_DIM;
    const float* zd = z + (size_t)d * OUT_DIM;

    const float zs0 = zs[lane],      zs1 = zs[lane + 32];
    const float zd0 = zd[lane],      zd1 = zd[lane + 32];

    // e = z_src . a[:64] + z_dst . a[64:]
    float p = zs0 * a_attn[lane]      + zs1 * a_attn[lane + 32]
            + zd0 * a_attn[64 + lane] + zd1 * a_attn[96 + lane];

    // wave32 butterfly reduction (5 steps — warpSize==32 on gfx1250)
#pragma unroll
    for (int off = 16; off > 0; off >>= 1)
        p += __shfl_xor(p, off, 32);

    const float e = (p >= 0.f) ? p : 0.01f * p;        // leaky_relu
    const float alpha = 1.f / (1.f + __expf(-e));      // sigmoid

    float* o = out + (size_t)d * OUT_DIM;
    atomicAdd(&o[lane],      alpha * zs0);             // global_atomic_add_f32
    atomicAdd(&o[lane + 32], alpha * zs1);
}

// ---------------------------------------------------------------------------
// launch
// ---------------------------------------------------------------------------
extern "C" void kernel_launch(void* const* d_in, const int* in_sizes, int n_in,
                              void* d_out, int out_size, void* d_ws, size_t ws_size,
                              hipStream_t stream) {
    const float* h      = (const float*)d_in[0];
    const int*   src    = (const int*)  d_in[1];
    const int*   dst    = (const int*)  d_in[2];
    const float* W_fc   = (const float*)d_in[3];
    const float* a_attn = (const float*)d_in[4];
    float*       out    = (float*)d_out;
    float*       z      = (float*)d_ws;         // [nNodes, 64] scratch (25.6 MB)

    const int nNodes = in_sizes[0] / IN_DIM;    // 100000
    const int nEdges = in_sizes[1];             // 1000000

    // 1) zero output accumulator (out_size floats)
    {
        const int n4 = out_size / 4;
        const int blocks = min((n4 + 255) / 256, 4096);
        gat_zero_out<<<blocks, 256, 0, stream>>>((float4*)out, n4);
    }

    // 2) z = h @ W_fc  (one wave per 16-row tile; 8 waves per 256-thread block)
    {
        const int nWaves = (nNodes + 15) / 16;          // 6250
        const int blocks = (nWaves + 7) / 8;
        gat_gemm_z<<<blocks, 256, 0, stream>>>(h, W_fc, z, nNodes);
    }

    // 3) edge attention + scatter-add (one wave per edge)
    {
        const int blocks = (nEdges + 7) / 8;            // 8 waves/block
        gat_edge<<<blocks, 256, 0, stream>>>(z, src, dst, a_attn, out, nEdges);
    }
}